// DySample_58110907515471
// MI455X (gfx1250) — compile-verified
//
#include <hip/hip_runtime.h>
#include <math.h>

typedef __attribute__((ext_vector_type(2))) float v2f;
typedef __attribute__((ext_vector_type(8))) float v8f;

#define B_  16
#define C_  64
#define H_  256
#define W_  256
#define HW_ (H_ * W_)   // 65536

// One block per (b, h): 256 threads = 8 waves (wave32).
// Phase 1: per-row offset GEMM via V_WMMA_F32_16X16X4_F32 -> px/py in LDS.
// Phase 2: per-pixel bilinear gather over all 64 channels.
__global__ __launch_bounds__(256) void dysample_fused_kernel(
    const float* __restrict__ x,      // [B,C,H,W]
    const float* __restrict__ w_off,  // [2,C]
    const float* __restrict__ b_off,  // [2]
    float* __restrict__ out)          // [B,C,H,W]
{
    __shared__ float sPX[W_];
    __shared__ float sPY[W_];

    const int blk  = blockIdx.x;
    const int b    = blk >> 8;      // blk / 256
    const int h    = blk & 255;     // blk % 256
    const int tid  = threadIdx.x;
    const int lane = tid & 31;
    const int wv   = tid >> 5;      // wave id 0..7
    const int n    = lane & 15;     // M (for A) / N (for B,C)
    const int half = lane >> 4;     // 0 or 1

    const float* xb = x + (size_t)b * C_ * HW_;

    // ---------------- Phase 1: offsets via WMMA fp32 ----------------
    // Preload all 16 B fragments branch-free (invariant across tiles).
    // B (4x16 fp32): row K striped across lanes; only columns N<2 are live.
    const float bmask = (n < 2) ? 1.0f : 0.0f;          // v_cndmask, no branch
    const int   wrow  = ((n < 2) ? n : 0) * C_;          // clamped lane address
    v2f Bfrag[16];
    #pragma unroll
    for (int ks = 0; ks < 16; ++ks) {
        const int cA = 4 * ks + 2 * half;
        const v2f wpair = *(const v2f*)(w_off + wrow + cA);  // global_load_b64
        Bfrag[ks].x = wpair.x * bmask;
        Bfrag[ks].y = wpair.y * bmask;
    }
    const float bias = b_off[(n < 2) ? n : 1];           // unconditional load

    for (int t = 0; t < 2; ++t) {
        const int pb = wv * 32 + t * 16;   // 16-pixel tile base in this row
        v8f acc = {};
        #pragma unroll
        for (int ks = 0; ks < 16; ++ks) {
            // A (16x4 fp32): lane half=0 holds K={0,1}, half=1 holds K={2,3}
            const int cA = 4 * ks + 2 * half;
            v2f A;
            A.x = xb[(size_t)cA       * HW_ + (size_t)h * W_ + pb + n];
            A.y = xb[(size_t)(cA + 1) * HW_ + (size_t)h * W_ + pb + n];
            acc = __builtin_amdgcn_wmma_f32_16x16x4_f32(
                false, A, false, Bfrag[ks], (short)0, acc, false, false);
        }
        // C/D layout: VGPR r, lane l -> M = r + 8*(l>>4), N = l&15.
        // Columns N=0 (x-offset) and N=1 (y-offset) carry the results.
        if (n < 2) {
            #pragma unroll
            for (int r = 0; r < 8; ++r) {
                const int   pix  = pb + r + 8 * half;
                const float off  = tanhf(acc[r] + bias) * 0.5f;
                const float base = -1.0f + (float)((n == 0) ? pix : h) * (2.0f / 255.0f);
                // px = ((g + 1)*W - 1)*0.5 with W == H == 256
                const float p = ((base + off + 1.0f) * 256.0f - 1.0f) * 0.5f;
                if (n == 0) sPX[pix] = p; else sPY[pix] = p;
            }
        }
    }
    __syncthreads();

    // ---------------- Phase 2: bilinear gather ----------------
    const float px  = sPX[tid];
    const float py  = sPY[tid];
    const float x0f = floorf(px);
    const float y0f = floorf(py);
    const float wx1 = px - x0f;
    const float wy1 = py - y0f;
    const int x0 = (int)x0f, y0 = (int)y0f;
    const int x1 = x0 + 1,   y1 = y0 + 1;

    const bool vx0 = (x0 >= 0) & (x0 < W_);
    const bool vx1 = (x1 >= 0) & (x1 < W_);
    const bool vy0 = (y0 >= 0) & (y0 < H_);
    const bool vy1 = (y1 >= 0) & (y1 < H_);

    const int xc0 = min(max(x0, 0), W_ - 1);
    const int xc1 = min(max(x1, 0), W_ - 1);
    const int yc0 = min(max(y0, 0), H_ - 1);
    const int yc1 = min(max(y1, 0), H_ - 1);

    float w00 = (1.0f - wx1) * (1.0f - wy1);
    float w01 = wx1 * (1.0f - wy1);
    float w10 = (1.0f - wx1) * wy1;
    float w11 = wx1 * wy1;
    if (!(vx0 && vy0)) w00 = 0.0f;   // padding_mode='zeros'
    if (!(vx1 && vy0)) w01 = 0.0f;
    if (!(vx0 && vy1)) w10 = 0.0f;
    if (!(vx1 && vy1)) w11 = 0.0f;

    const int i00 = yc0 * W_ + xc0;
    const int i01 = yc0 * W_ + xc1;
    const int i10 = yc1 * W_ + xc0;
    const int i11 = yc1 * W_ + xc1;

    float* ob = out + (size_t)b * C_ * HW_ + (size_t)h * W_ + tid;
    #pragma unroll 8
    for (int c = 0; c < C_; ++c) {
        const float* xc = xb + (size_t)c * HW_;
        const float v = w00 * xc[i00] + w01 * xc[i01]
                      + w10 * xc[i10] + w11 * xc[i11];
        ob[(size_t)c * HW_] = v;
    }
}

extern "C" void kernel_launch(void* const* d_in, const int* in_sizes, int n_in,
                              void* d_out, int out_size, void* d_ws, size_t ws_size,
                              hipStream_t stream) {
    (void)in_sizes; (void)n_in; (void)d_ws; (void)ws_size; (void)out_size;
    const float* x     = (const float*)d_in[0];
    const float* w_off = (const float*)d_in[1];
    const float* b_off = (const float*)d_in[2];
    float*       out   = (float*)d_out;

    dysample_fused_kernel<<<dim3(B_ * H_), dim3(256), 0, stream>>>(x, w_off, b_off, out);
}